// LinearDistributed_54820962566194
// MI455X (gfx1250) — compile-verified
//
#include <hip/hip_runtime.h>

typedef __attribute__((ext_vector_type(2))) float v2f;
typedef __attribute__((ext_vector_type(8))) float v8f;

#define BATCH   512
#define SLIST   512
#define DIM     256
#define NHOPS   3
#define TILE_S  16
#define WAVES   4   // waves per block
#define BLOCKT  (WAVES * 32)

__global__ __launch_bounds__(BLOCKT, 1)
void galaxc_shortlist_wmma(const float* __restrict__ embed,      // [B, 3*256]
                           const int*   __restrict__ shortlist,  // [B, S]
                           const float* __restrict__ weight,     // [L, 256]
                           const float* __restrict__ bias,       // [L]
                           const float* __restrict__ att,        // [L, 3]
                           float* __restrict__ out)              // [B, S]
{
    __shared__ float lds_emb[NHOPS * DIM];              // 3 KB, shared per block
    __shared__ float lds_d[WAVES][TILE_S][TILE_S];      // 4 KB, per-wave D tile

    const int tid  = threadIdx.x;
    const int wave = tid >> 5;
    const int lane = tid & 31;
    const int n    = lane & 15;   // A: row m / B: col n / D: col n
    const int hi   = lane >> 4;   // half-wave select (K sub-pair)

    const int b    = blockIdx.x >> 3;          // 8 s-groups per batch row
    const int s0   = (blockIdx.x & 7) * (WAVES * TILE_S) + wave * TILE_S;

    // ---- stage this batch row's 3 hop embeddings into LDS (coalesced) ----
    for (int i = tid; i < NHOPS * DIM; i += BLOCKT)
        lds_emb[i] = embed[b * (NHOPS * DIM) + i];
    __syncthreads();

    // ---- gather shortlist index for this lane's tile row ----
    const int l = shortlist[b * SLIST + s0 + n];
    const float* wrow = weight + (long long)l * DIM;

    // B-fragment source: hop n (columns >= NHOPS are zeroed via sel)
    const float sel   = (n < NHOPS) ? 1.0f : 0.0f;
    const int   bbase = ((n < NHOPS) ? n : 0) * DIM + 2 * hi;

    // ---- K loop: 64 steps of V_WMMA_F32_16X16X4_F32, f32 accumulate ----
    v8f c = {0.f, 0.f, 0.f, 0.f, 0.f, 0.f, 0.f, 0.f};
#pragma unroll 8
    for (int kc = 0; kc < DIM; kc += 4) {
        // A-frag: one b64 gather per lane (full-width streaming of 16 rows)
        v2f a = *(const v2f*)(wrow + kc + 2 * hi);
        // B-frag: b64 from LDS-staged embeddings
        v2f bb = *(const v2f*)(&lds_emb[bbase + kc]);
        bb.x *= sel;
        bb.y *= sel;
        c = __builtin_amdgcn_wmma_f32_16x16x4_f32(
                /*neg_a=*/false, a, /*neg_b=*/false, bb,
                /*c_mod=*/(short)0, c, /*reuse_a=*/false, /*reuse_b=*/false);
    }

    // ---- spill D tile to LDS to transpose lane ownership ----
#pragma unroll
    for (int r = 0; r < 8; ++r)
        lds_d[wave][r + 8 * hi][n] = c[r];

    asm volatile("s_wait_dscnt 0" ::: "memory");
    __builtin_amdgcn_wave_barrier();

    // ---- lanes 0..15: softmax(att[l]) combine + bias, store ----
    if (lane < TILE_S) {
        const int m = lane;                 // n == m here, so `l` matches row m
        const float d0 = lds_d[wave][m][0];
        const float d1 = lds_d[wave][m][1];
        const float d2 = lds_d[wave][m][2];

        const float* ar = att + (long long)l * NHOPS;
        const float a0 = ar[0], a1 = ar[1], a2 = ar[2];
        const float mx = fmaxf(a0, fmaxf(a1, a2));
        const float e0 = __expf(a0 - mx);
        const float e1 = __expf(a1 - mx);
        const float e2 = __expf(a2 - mx);
        const float inv = 1.0f / (e0 + e1 + e2);

        out[b * SLIST + s0 + m] = (d0 * e0 + d1 * e1 + d2 * e2) * inv + bias[l];
    }
}

extern "C" void kernel_launch(void* const* d_in, const int* in_sizes, int n_in,
                              void* d_out, int out_size, void* d_ws, size_t ws_size,
                              hipStream_t stream) {
    (void)in_sizes; (void)n_in; (void)d_ws; (void)ws_size; (void)out_size;
    const float* embed     = (const float*)d_in[0];  // [512, 768]
    const int*   shortlist = (const int*)  d_in[1];  // [512, 512]
    const float* weight    = (const float*)d_in[2];  // [500000, 256]
    const float* bias      = (const float*)d_in[3];  // [500000]
    const float* att       = (const float*)d_in[4];  // [500000, 3]
    float*       out       = (float*)d_out;          // [512, 512]

    // 512 batches * (512 / 64) s-groups = 4096 blocks of 128 threads (4 waves)
    dim3 grid(BATCH * (SLIST / (WAVES * TILE_S)));
    dim3 block(BLOCKT);
    galaxc_shortlist_wmma<<<grid, block, 0, stream>>>(embed, shortlist, weight,
                                                      bias, att, out);
}